// Gemma3nTextAltUp_2783138808559
// MI455X (gfx1250) — compile-verified
//
#include <hip/hip_runtime.h>
#include <cstdint>

typedef __attribute__((ext_vector_type(2))) float v2f;
typedef __attribute__((ext_vector_type(8))) float v8f;

#define HID        2048
#define NTOK_BLK   16
#define THREADS    256
#define EPSF       1e-6f

__global__ __launch_bounds__(THREADS) void altup_fused_kernel(
    const float* __restrict__ hs,    // [4, NTOK, H]
    const float* __restrict__ act,   // [NTOK, H]
    const float* __restrict__ rnw,   // [H]
    const float* __restrict__ wr,    // [4, H] row-major
    const float* __restrict__ wp,    // [16, 4] row-major
    const float* __restrict__ wc,    // [4, 4]  row-major
    const float* __restrict__ osc,   // [H]
    float* __restrict__ out,         // [NTOK, H]
    int ntok)
{
  // LDS: [0,2048)=rnw ; [2048,10240)=W_router rows 0..3 ; [10240,12288)=out_scale
  __shared__ float s_wgt[6 * HID];
  __shared__ float s_m [NTOK_BLK][4];
  __shared__ float s_m2[NTOK_BLK][4];
  __shared__ float s_C [NTOK_BLK][4];
  __shared__ float s_cf[NTOK_BLK];

  const int tid = threadIdx.x;

  // ---- CDNA5 async global->LDS copy of the weight vectors (48 KB) ----
  {
    #pragma unroll
    for (int i = 0; i < 12; ++i) {
      int f = i * (THREADS * 4) + tid * 4;               // float index
      const float* src;
      if (f < 2048)        src = rnw + f;
      else if (f < 10240)  src = wr  + (f - 2048);
      else                 src = osc + (f - 10240);
      unsigned lds_b = (unsigned)(uintptr_t)(&s_wgt[f]); // LDS byte offset
      unsigned long long ga = (unsigned long long)(uintptr_t)src;
      asm volatile("global_load_async_to_lds_b128 %0, %1, off"
                   :: "v"(lds_b), "v"(ga) : "memory");
    }
    asm volatile("s_wait_asynccnt 0" ::: "memory");
  }
  __syncthreads();

  const int tok_l = tid >> 4;                 // 0..15 (two tokens per wave32)
  const int sub   = tid & 15;
  const long long tok   = (long long)blockIdx.x * NTOK_BLK + tok_l;
  const long long plane = (long long)ntok * HID;
  const float* __restrict__ hs0 = hs + tok * HID;
  const float* __restrict__ h1  = hs0 + plane;
  const float* __restrict__ h2  = hs0 + 2 * plane;
  const float* __restrict__ h3  = hs0 + 3 * plane;
  const float* __restrict__ acp = act + tok * HID;

  // ---- Phase 1: fused reductions over H (ss + 4 router dots, x2 streams) ----
  float ss0 = 0.f, d0 = 0.f, d1 = 0.f, d2 = 0.f, d3 = 0.f;
  float ss2 = 0.f, e0 = 0.f, e1 = 0.f, e2 = 0.f, e3 = 0.f;

  #pragma unroll 4
  for (int k = 0; k < HID / 64; ++k) {
    const int h = k * 64 + sub * 4;
    float4 x  = *(const float4*)(hs0 + h);
    float4 y  = *(const float4*)(acp + h);
    float4 w  = *(const float4*)(&s_wgt[h]);
    float4 w0 = *(const float4*)(&s_wgt[2048 + h]);
    float4 w1 = *(const float4*)(&s_wgt[4096 + h]);
    float4 w2 = *(const float4*)(&s_wgt[6144 + h]);
    float4 w3 = *(const float4*)(&s_wgt[8192 + h]);
    if ((k & 3) == 0) {   // warm caches for the a=1..3 slices used in phase 2
      __builtin_prefetch(h1 + h, 0, 1);
      __builtin_prefetch(h2 + h, 0, 1);
      __builtin_prefetch(h3 + h, 0, 1);
    }
    ss0 += x.x*x.x + x.y*x.y + x.z*x.z + x.w*x.w;
    ss2 += y.x*y.x + y.y*y.y + y.z*y.z + y.w*y.w;
    float t0 = x.x*w.x, t1 = x.y*w.y, t2 = x.z*w.z, t3 = x.w*w.w;
    d0 += t0*w0.x + t1*w0.y + t2*w0.z + t3*w0.w;
    d1 += t0*w1.x + t1*w1.y + t2*w1.z + t3*w1.w;
    d2 += t0*w2.x + t1*w2.y + t2*w2.z + t3*w2.w;
    d3 += t0*w3.x + t1*w3.y + t2*w3.z + t3*w3.w;
    float u0 = y.x*w.x, u1 = y.y*w.y, u2 = y.z*w.z, u3 = y.w*w.w;
    e0 += u0*w0.x + u1*w0.y + u2*w0.z + u3*w0.w;
    e1 += u0*w1.x + u1*w1.y + u2*w1.z + u3*w1.w;
    e2 += u0*w2.x + u1*w2.y + u2*w2.z + u3*w2.w;
    e3 += u0*w3.x + u1*w3.y + u2*w3.z + u3*w3.w;
  }

  // reduce across the 16 lanes owning this token (stays inside half-wave)
  #pragma unroll
  for (int msk = 1; msk <= 8; msk <<= 1) {
    ss0 += __shfl_xor(ss0, msk, 32);  ss2 += __shfl_xor(ss2, msk, 32);
    d0  += __shfl_xor(d0,  msk, 32);  e0  += __shfl_xor(e0,  msk, 32);
    d1  += __shfl_xor(d1,  msk, 32);  e1  += __shfl_xor(e1,  msk, 32);
    d2  += __shfl_xor(d2,  msk, 32);  e2  += __shfl_xor(e2,  msk, 32);
    d3  += __shfl_xor(d3,  msk, 32);  e3  += __shfl_xor(e3,  msk, 32);
  }

  const float invH = 1.0f / (float)HID;
  float r1 = __frsqrt_rn(ss0 * invH + EPSF) * invH;
  float r2 = __frsqrt_rn(ss2 * invH + EPSF) * invH;
  if (sub == 0) {
    s_m [tok_l][0] = tanhf(d0 * r1); s_m [tok_l][1] = tanhf(d1 * r1);
    s_m [tok_l][2] = tanhf(d2 * r1); s_m [tok_l][3] = tanhf(d3 * r1);
    s_m2[tok_l][0] = tanhf(e0 * r2); s_m2[tok_l][1] = tanhf(e1 * r2);
    s_m2[tok_l][2] = tanhf(e2 * r2); s_m2[tok_l][3] = tanhf(e3 * r2);
  }
  __syncthreads();

  // ---- WMMA: C = m @ Wp^T  (M=16 tokens, K=4, N=16), coefs = m2 @ Wc^T ----
  if (tid < 32) {
    const int lane  = tid;
    const int row16 = lane & 15;
    const bool hi   = lane >= 16;
    const int klo = hi ? 2 : 0, khi = hi ? 3 : 1;   // 16x4 f32 A layout: K0/K2, K1/K3
    v2f a1, a2, b1, b2;
    a1.x = s_m [row16][klo]; a1.y = s_m [row16][khi];
    a2.x = s_m2[row16][klo]; a2.y = s_m2[row16][khi];
    b1.x = wp[row16 * 4 + klo];                      // B[k][n] = Wp[n][k]
    b1.y = wp[row16 * 4 + khi];
    b2.x = (row16 < 4) ? wc[row16 * 4 + klo] : 0.f;  // B[k][n] = Wc[n][k], n<4
    b2.y = (row16 < 4) ? wc[row16 * 4 + khi] : 0.f;
    v8f z = {};
    v8f D1 = __builtin_amdgcn_wmma_f32_16x16x4_f32(false, a1, false, b1,
                                                   (short)0, z, false, false);
    v8f D2 = __builtin_amdgcn_wmma_f32_16x16x4_f32(false, a2, false, b2,
                                                   (short)0, z, false, false);
    const int col = lane & 15;
    if (col < 4) {                    // D layout: vgpr r -> row r (lo) / 8+r (hi)
      const int rb = hi ? 8 : 0;
      #pragma unroll
      for (int r = 0; r < 8; ++r) s_C[rb + r][col] = D1[r];
      if (col == 0) {
        #pragma unroll
        for (int r = 0; r < 8; ++r) s_cf[rb + r] = D2[r] + 1.0f;
      }
    }
  }
  __syncthreads();

  // ---- Phase 2: fused predict/correct/scale, streaming pass ----
  const float C00 = s_C[tok_l][0] + 1.0f;   // + residual hs[0]
  const float C01 = s_C[tok_l][1];
  const float C02 = s_C[tok_l][2];
  const float C03 = s_C[tok_l][3];
  const float cf  = s_cf[tok_l];
  float* __restrict__ op = out + tok * HID;

  #pragma unroll 4
  for (int k = 0; k < HID / 64; ++k) {
    const int h = k * 64 + sub * 4;
    float4 x0 = *(const float4*)(hs0 + h);   // L2 hit (just streamed)
    float4 x1 = *(const float4*)(h1 + h);
    float4 x2 = *(const float4*)(h2 + h);
    float4 x3 = *(const float4*)(h3 + h);
    float4 y  = *(const float4*)(acp + h);   // L2 hit
    float4 sc = *(const float4*)(&s_wgt[10240 + h]);
    float4 o;
    {
      float p = x0.x*C00 + x1.x*C01 + x2.x*C02 + x3.x*C03;
      o.x = ((y.x - p) * cf + p) * sc.x;
    }
    {
      float p = x0.y*C00 + x1.y*C01 + x2.y*C02 + x3.y*C03;
      o.y = ((y.y - p) * cf + p) * sc.y;
    }
    {
      float p = x0.z*C00 + x1.z*C01 + x2.z*C02 + x3.z*C03;
      o.z = ((y.z - p) * cf + p) * sc.z;
    }
    {
      float p = x0.w*C00 + x1.w*C01 + x2.w*C02 + x3.w*C03;
      o.w = ((y.w - p) * cf + p) * sc.w;
    }
    *(float4*)(op + h) = o;
  }
}

extern "C" void kernel_launch(void* const* d_in, const int* in_sizes, int n_in,
                              void* d_out, int out_size, void* d_ws, size_t ws_size,
                              hipStream_t stream) {
  const float* hs  = (const float*)d_in[0];  // [4,B,T,H]
  const float* act = (const float*)d_in[1];  // [B,T,H]
  const float* rnw = (const float*)d_in[2];  // [H]
  const float* wr  = (const float*)d_in[3];  // [4,H]
  const float* wp  = (const float*)d_in[4];  // [16,4]
  const float* wc  = (const float*)d_in[5];  // [4,4]
  const float* osc = (const float*)d_in[6];  // [H]
  float* out = (float*)d_out;

  const int ntok = in_sizes[1] / HID;        // B*T = 16384
  const int nblk = ntok / NTOK_BLK;          // 1024 blocks
  altup_fused_kernel<<<nblk, THREADS, 0, stream>>>(hs, act, rnw, wr, wp, wc,
                                                   osc, out, ntok);
}